// Mymodel_29575144800973
// MI455X (gfx1250) — compile-verified
//
#include <hip/hip_runtime.h>
#include <stdint.h>

typedef float    v4f __attribute__((ext_vector_type(4)));
typedef uint32_t v4u __attribute__((ext_vector_type(4)));
typedef uint32_t v8u __attribute__((ext_vector_type(8)));

#define M3D 17
#define N1D 64
#define TEX3 (M3D*M3D*M3D)            /* 4913 texels */
#define LUT3_FLOATS (TEX3*3)          /* 14739 dwords = 58,956 B raw */
#define PAD_DWORDS (TEX3*4)           /* 19652 dwords = 78,608 B padded (float4/texel) */
#define SMEM_DWORDS (PAD_DWORDS + 3*N1D)
#define SMEM_BYTES  (SMEM_DWORDS*4)   /* 79,376 B */
#define HW   (1024*1024)
#define HW4  (HW/4)                   /* 262144 float4 groups per plane */
#define NBATCH 4

__device__ __forceinline__ float lerpf(float a, float b, float t) {
    return fmaf(t, b - a, a);
}
__device__ __forceinline__ v4f lerp4(v4f a, v4f b, float t) {
    return a + t * (b - a);          // contracts to v_pk_fma_f32 pairs
}

// 1D LUT: xs = v*63, i0 = clamp(floor,0,62), lerp, clip to [0,1]
__device__ __forceinline__ float apply1d_clip(float v, const float* __restrict__ t) {
    float xs = v * (float)(N1D - 1);
    int i = (int)floorf(xs);
    i = i < 0 ? 0 : (i > N1D - 2 ? N1D - 2 : i);
    float f = xs - (float)i;
    float y = lerpf(t[i], t[i + 1], f);
    return fminf(fmaxf(y, 0.0f), 1.0f);
}

__global__ __launch_bounds__(256) void lut_apply_kernel(
    const float* __restrict__ x,      // (4,3,1024,1024)
    const float* __restrict__ lut1d,  // (3,64)
    const float* __restrict__ lut3d,  // (17,17,17,3)
    float* __restrict__ out)          // (4,3,1024,1024)
{
    extern __shared__ __attribute__((aligned(16))) float smem[];
    float* s3p = smem;                 // padded LUT: TEX3 x float4
    float* s1  = smem + PAD_DWORDS;    // 3 x 64 floats

    // tiny 1D LUT: cooperative load (192 dwords)
    if (threadIdx.x < 3 * N1D) s1[threadIdx.x] = lut1d[threadIdx.x];

    // 3D LUT: one TDM DMA (Global -> LDS tail), issued by wave 0 only
    // (scalar SGPR branch, so waves 1..7 never reach the tensor op).
    unsigned wave = __builtin_amdgcn_readfirstlane(threadIdx.x >> 5);
    if (wave == 0) {
        uint64_t ga = (uint64_t)(uintptr_t)lut3d;
        uint32_t la = (uint32_t)(uintptr_t)(&smem[TEX3]);  // raw data at dword offset TEX3

        // D# group 0: count=1 | lds_addr | global_addr[56:0] | type=2
        v4u g0;
        g0[0] = 1u;
        g0[1] = la;
        g0[2] = (uint32_t)ga;
        g0[3] = ((uint32_t)(ga >> 32) & 0x01FFFFFFu) | (2u << 30);

        // D# group 1: data_size=4B; one row of 14739 dwords (fits 16-bit tile_dim0)
        v8u g1;
        g1[0] = (2u << 16);                                  // wg_mask=0, data_size=2 (4B)
        g1[1] = ((uint32_t)LUT3_FLOATS & 0xFFFFu) << 16;     // tensor_dim0[15:0]
        g1[2] = ((uint32_t)LUT3_FLOATS >> 16) | (1u << 16);  // tensor_dim0[31:16] | tensor_dim1=1
        g1[3] = ((uint32_t)LUT3_FLOATS) << 16;               // tile_dim0 = 14739
        g1[4] = 0u;                                          // tile_dim1/2 unused
        g1[5] = (uint32_t)LUT3_FLOATS;                       // tensor_dim0_stride[31:0]
        g1[6] = 0u;
        g1[7] = 0u;

        asm volatile("tensor_load_to_lds %0, %1" :: "s"(g0), "s"(g1) : "memory");
        __builtin_amdgcn_s_wait_tensorcnt(0);
    }
    __syncthreads();

    // In-place expansion: raw [TEX3, TEX3+14739) -> padded float4 [0, 19652).
    // Phase p: reads >= TEX3+768p, writes <= 1024p+1023: disjoint across phases;
    // within a phase the read / barrier / write split protects the overlap.
    for (int base_i = 0; base_i < TEX3; base_i += 256) {
        int i = base_i + (int)threadIdx.x;
        float a = 0.f, b = 0.f, c = 0.f;
        bool act = (i < TEX3);
        if (act) {
            a = smem[TEX3 + 3 * i + 0];
            b = smem[TEX3 + 3 * i + 1];
            c = smem[TEX3 + 3 * i + 2];
        }
        __syncthreads();
        if (act) {
            v4f t = {a, b, c, 0.0f};
            *((v4f*)s3p + i) = t;
        }
    }
    __syncthreads();

    const v4f* __restrict__ xv = (const v4f*)x;
    v4f* __restrict__ ov = (v4f*)out;
    const v4f* __restrict__ lut = (const v4f*)s3p;

    const int NG = NBATCH * HW4;
    const int stride = (int)(gridDim.x * blockDim.x);
    for (int gidx = (int)(blockIdx.x * blockDim.x + threadIdx.x); gidx < NG; gidx += stride) {
        int bidx = gidx >> 18;                   // / HW4
        int p    = gidx & (HW4 - 1);
        size_t base = (size_t)(bidx * 3) * HW4 + (size_t)p;

        // streaming, non-temporal: keep caches for the LUT
        v4f R = __builtin_nontemporal_load(xv + base);
        v4f G = __builtin_nontemporal_load(xv + base + HW4);
        v4f B = __builtin_nontemporal_load(xv + base + 2 * HW4);

        v4f Ro, Go, Bo;
#pragma unroll
        for (int j = 0; j < 4; ++j) {
            // stage 1: per-channel 1D LUT + clip
            float r = apply1d_clip(R[j], s1 + 0 * N1D);
            float g = apply1d_clip(G[j], s1 + 1 * N1D);
            float b = apply1d_clip(B[j], s1 + 2 * N1D);

            // stage 2: trilinear 3D LUT
            float xr = r * (float)(M3D - 1);
            float xg = g * (float)(M3D - 1);
            float xb = b * (float)(M3D - 1);
            int ir = (int)floorf(xr); ir = ir < 0 ? 0 : (ir > M3D - 2 ? M3D - 2 : ir);
            int ig = (int)floorf(xg); ig = ig < 0 ? 0 : (ig > M3D - 2 ? M3D - 2 : ig);
            int ib = (int)floorf(xb); ib = ib < 0 ? 0 : (ib > M3D - 2 ? M3D - 2 : ib);
            float fr = xr - (float)ir;
            float fg = xg - (float)ig;
            float fb = xb - (float)ib;

            const v4f* pc = lut + ((ir * M3D + ig) * M3D + ib);
            // 8 corners: single address, ds_load_b128 with immediate offsets
            v4f c000 = pc[0];
            v4f c001 = pc[1];
            v4f c010 = pc[M3D];
            v4f c011 = pc[M3D + 1];
            v4f c100 = pc[M3D * M3D];
            v4f c101 = pc[M3D * M3D + 1];
            v4f c110 = pc[M3D * M3D + M3D];
            v4f c111 = pc[M3D * M3D + M3D + 1];

            v4f c00 = lerp4(c000, c001, fb);
            v4f c01 = lerp4(c010, c011, fb);
            v4f c10 = lerp4(c100, c101, fb);
            v4f c11 = lerp4(c110, c111, fb);
            v4f c0  = lerp4(c00, c01, fg);
            v4f c1  = lerp4(c10, c11, fg);
            v4f res = lerp4(c0, c1, fr);

            Ro[j] = res[0]; Go[j] = res[1]; Bo[j] = res[2];
        }

        __builtin_nontemporal_store(Ro, ov + base);
        __builtin_nontemporal_store(Go, ov + base + HW4);
        __builtin_nontemporal_store(Bo, ov + base + 2 * HW4);
    }
}

extern "C" void kernel_launch(void* const* d_in, const int* in_sizes, int n_in,
                              void* d_out, int out_size, void* d_ws, size_t ws_size,
                              hipStream_t stream) {
    (void)in_sizes; (void)n_in; (void)out_size; (void)d_ws; (void)ws_size;
    const float* x     = (const float*)d_in[0];
    const float* lut1d = (const float*)d_in[1];
    const float* lut3d = (const float*)d_in[2];
    float* out = (float*)d_out;

    // 1024 blocks x 256 threads (8 wave32s); 79.4 KB dynamic LDS -> 4 blocks/WGP.
    lut_apply_kernel<<<dim3(1024), dim3(256), SMEM_BYTES, stream>>>(x, lut1d, lut3d, out);
}